// GNNModel_32925219291768
// MI455X (gfx1250) — compile-verified
//
#include <hip/hip_runtime.h>
#include <math.h>

typedef float v2f __attribute__((ext_vector_type(2)));
typedef float v8f __attribute__((ext_vector_type(8)));

#define DIVUP(a,b) (((a)+(b)-1)/(b))

static const int Nn = 50000;   // nodes
static const int Ee = 800000;  // edges
static const int Gg = 1024;    // graphs

// ---------------------------------------------------------------------------
// float atomic max via int/uint monotonicity trick (works for all finite+inf)
// ---------------------------------------------------------------------------
__device__ __forceinline__ void atomicMaxF(float* addr, float val) {
  if (val >= 0.0f) {
    atomicMax((int*)addr, __float_as_int(val));
  } else {
    atomicMin((unsigned int*)addr, __float_as_uint(val));
  }
}

// ---------------------------------------------------------------------------
// fp32 WMMA GEMM:  C[M,N] = act(A[M,K] @ B[K,N] + bias)
// Register-blocked: one 16x64 output strip per wave32 (4 accumulators),
// so each K-step is 1 A-fragment + 8 B elements feeding 4 WMMAs.
// Requires: M%16==0, N%64==0, K%4==0 (true for all uses here).
// A-frag (ISA 16x4 layout): lane L, vgpr v -> m = L%16, k = 2*(L/16)+v
// B-frag (4x16, mirrored):  lane L, vgpr v -> n = L%16, k = 2*(L/16)+v
// C/D   (16x16):            lane L, vgpr v -> n = L%16, m = v + 8*(L/16)
// ---------------------------------------------------------------------------
__global__ void gemm_f32_wmma(const float* __restrict__ A, const float* __restrict__ B,
                              const float* __restrict__ bias, float* __restrict__ C,
                              int M, int N, int K, int lda, int ldb, int ldc, int relu_act) {
  int wave = (blockIdx.x * blockDim.x + threadIdx.x) >> 5;
  int lane = threadIdx.x & 31;
  int Mt = M >> 4, Nt4 = N >> 6;
  if (wave >= Mt * Nt4) return;           // uniform per wave -> EXEC stays all-1s
  int m0 = (wave / Nt4) << 4;
  int n0 = (wave % Nt4) << 6;
  int r  = lane & 15;                     // A row / B,C col within tile
  int kk = (lane >> 4) << 1;              // K sub-offset: 0 or 2
  v8f acc0 = {0.f,0.f,0.f,0.f,0.f,0.f,0.f,0.f};
  v8f acc1 = acc0, acc2 = acc0, acc3 = acc0;
  const float* arow = A + (size_t)(m0 + r) * lda;
  for (int k0 = 0; k0 < K; k0 += 4) {
    v2f a;
    a[0] = arow[k0 + kk];
    a[1] = arow[k0 + kk + 1];
    const float* b0 = B + (size_t)(k0 + kk) * ldb + n0 + r;       // k = kk
    const float* b1 = b0 + ldb;                                   // k = kk+1
    v2f bA = { b0[0],  b1[0]  };
    v2f bB = { b0[16], b1[16] };
    v2f bC = { b0[32], b1[32] };
    v2f bD = { b0[48], b1[48] };
    acc0 = __builtin_amdgcn_wmma_f32_16x16x4_f32(false, a, false, bA, (short)0, acc0, false, false);
    acc1 = __builtin_amdgcn_wmma_f32_16x16x4_f32(false, a, false, bB, (short)0, acc1, false, false);
    acc2 = __builtin_amdgcn_wmma_f32_16x16x4_f32(false, a, false, bC, (short)0, acc2, false, false);
    acc3 = __builtin_amdgcn_wmma_f32_16x16x4_f32(false, a, false, bD, (short)0, acc3, false, false);
  }
  int mh = (lane >> 4) << 3;              // 0 or 8
#pragma unroll
  for (int t = 0; t < 4; ++t) {
    v8f acc = (t == 0) ? acc0 : (t == 1) ? acc1 : (t == 2) ? acc2 : acc3;
    int nc = n0 + t * 16 + r;
    float bv = bias ? bias[nc] : 0.0f;
#pragma unroll
    for (int v = 0; v < 8; ++v) {
      float val = acc[v] + bv;
      if (relu_act) val = fmaxf(val, 0.0f);
      C[(size_t)(m0 + mh + v) * ldc + nc] = val;
    }
  }
}

// ---------------------------------------------------------------------------
// Small utility kernels
// ---------------------------------------------------------------------------
__global__ void k_zero(float* __restrict__ p, int n) {
  int i = blockIdx.x * blockDim.x + threadIdx.x;
  if (i < n) p[i] = 0.0f;
}

__global__ void k_init_out(float* __restrict__ outp, const float* __restrict__ bias,
                           int total, int OUT) {
  int i = blockIdx.x * blockDim.x + threadIdx.x;
  if (i < total) outp[i] = bias[i % OUT];
}

__global__ void k_init_nh(float* __restrict__ amax, float* __restrict__ denom, int n) {
  int i = blockIdx.x * blockDim.x + threadIdx.x;
  if (i < n) { amax[i] = -__builtin_inff(); denom[i] = 0.0f; }
}

// alpha_src[n,h] = sum_c h[n,h,c]*a_src[h,c]  (same for dst)
__global__ void k_alpha_nodes(const float* __restrict__ Hf, const float* __restrict__ a_s,
                              const float* __restrict__ a_d, float* __restrict__ aS,
                              float* __restrict__ aD, int N, int H, int C) {
  int idx = blockIdx.x * blockDim.x + threadIdx.x;
  if (idx >= N * H) return;
  int n = idx / H, h = idx % H;
  const float* hp  = Hf + (size_t)n * H * C + (size_t)h * C;
  const float* asp = a_s + (size_t)h * C;
  const float* adp = a_d + (size_t)h * C;
  float s = 0.f, d = 0.f;
  for (int c = 0; c < C; ++c) { float v = hp[c]; s += v * asp[c]; d += v * adp[c]; }
  aS[idx] = s; aD[idx] = d;
}

// M[k,h] = sum_c We[k, h*C+c] * a_edge[h,c]   (7 x H, collapses the he tensor)
__global__ void k_mproj(const float* __restrict__ We, const float* __restrict__ a_e,
                        float* __restrict__ Mp, int H, int C) {
  int idx = threadIdx.x;
  if (idx >= 7 * H) return;
  int k = idx / H, h = idx % H;
  float s = 0.f;
  for (int c = 0; c < C; ++c) s += We[(size_t)k * H * C + (size_t)h * C + c] * a_e[(size_t)h * C + c];
  Mp[idx] = s;
}

// alpha_e[e,h] = sum_k edge_attr[e,k] * M[k,h]
__global__ void k_alpha_edge(const float* __restrict__ ea, const float* __restrict__ Mp,
                             float* __restrict__ alE, int E, int H) {
  int idx = blockIdx.x * blockDim.x + threadIdx.x;
  if (idx >= E * H) return;
  int e = idx / H, h = idx % H;
  float s = 0.f;
#pragma unroll
  for (int k = 0; k < 7; ++k) s += ea[(size_t)e * 7 + k] * Mp[k * H + h];
  alE[idx] = s;
}

// alpha = leaky_relu(aS[src]+aD[dst]+alpha_e, 0.2); segment max over dst
__global__ void k_edge_lrelu_max(const int* __restrict__ src, const int* __restrict__ dst,
                                 const float* __restrict__ aS, const float* __restrict__ aD,
                                 float* __restrict__ alE, float* __restrict__ amax,
                                 int E, int H) {
  int idx = blockIdx.x * blockDim.x + threadIdx.x;
  if (idx >= E * H) return;
  int e = idx / H, h = idx % H;
  int s = src[e], d = dst[e];
  float v = aS[s * H + h] + aD[d * H + h] + alE[idx];
  v = (v > 0.0f) ? v : 0.2f * v;
  alE[idx] = v;
  atomicMaxF(&amax[d * H + h], v);
}

// ex = exp(alpha - amax[dst]); segment sum over dst
__global__ void k_edge_exp(const int* __restrict__ dst, float* __restrict__ alE,
                           const float* __restrict__ amax, float* __restrict__ denom,
                           int E, int H) {
  int idx = blockIdx.x * blockDim.x + threadIdx.x;
  if (idx >= E * H) return;
  int e = idx / H, h = idx % H;
  int d = dst[e];
  float v = __expf(alE[idx] - amax[d * H + h]);
  alE[idx] = v;
  atomicAdd(&denom[d * H + h], v);
}

// out[dst, h*C+c] += h[src, h*C+c] * (ex / (denom[dst]+eps)); one block per edge
__global__ void k_edge_msg(const int* __restrict__ src, const int* __restrict__ dst,
                           const float* __restrict__ Hf, const float* __restrict__ alE,
                           const float* __restrict__ denom, float* __restrict__ outp,
                           int H, int C) {
  int e = blockIdx.x;
  int t = threadIdx.x;               // 0 .. H*C-1
  int h = t / C;
  int s = src[e], d = dst[e];
  float w = alE[(size_t)e * H + h] / (denom[(size_t)d * H + h] + 1e-16f);
  atomicAdd(&outp[(size_t)d * H * C + t], Hf[(size_t)s * H * C + t] * w);
}

// BatchNorm (training-mode biased stats) + ReLU, in place
__global__ void k_bn_stats(const float* __restrict__ X, float* __restrict__ sum,
                           float* __restrict__ sq, int N, int OUT, int rpb) {
  int t  = threadIdx.x;              // OUT threads, one channel each
  int r0 = blockIdx.x * rpb;
  int r1 = r0 + rpb; if (r1 > N) r1 = N;
  float s = 0.f, q = 0.f;
  for (int r = r0; r < r1; ++r) { float v = X[(size_t)r * OUT + t]; s += v; q += v * v; }
  atomicAdd(&sum[t], s);
  atomicAdd(&sq[t], q);
}

__global__ void k_bn_apply(float* __restrict__ X, const float* __restrict__ sum,
                           const float* __restrict__ sq, const float* __restrict__ g,
                           const float* __restrict__ b, int N, int OUT) {
  int i = blockIdx.x * blockDim.x + threadIdx.x;
  if (i >= N * OUT) return;
  int j = i % OUT;
  float inv_n = 1.0f / (float)N;
  float m   = sum[j] * inv_n;
  float var = sq[j] * inv_n - m * m;
  float v = (X[i] - m) * rsqrtf(var + 1e-5f) * g[j] + b[j];
  X[i] = fmaxf(v, 0.0f);
}

// Global mean pool
__global__ void k_cnt(const int* __restrict__ batch, float* __restrict__ cnt, int N) {
  int i = blockIdx.x * blockDim.x + threadIdx.x;
  if (i < N) atomicAdd(&cnt[batch[i]], 1.0f);
}

__global__ void k_pool_scatter(const float* __restrict__ X, const int* __restrict__ batch,
                               float* __restrict__ pool, int N) {
  int i = blockIdx.x * blockDim.x + threadIdx.x;
  if (i >= N * 64) return;
  int n = i >> 6, j = i & 63;
  atomicAdd(&pool[(size_t)batch[n] * 64 + j], X[i]);
}

__global__ void k_pool_final(const float* __restrict__ pool, const float* __restrict__ cnt,
                             float* __restrict__ xg, float* __restrict__ comb, int G) {
  int i = blockIdx.x * blockDim.x + threadIdx.x;
  if (i >= G * 64) return;
  int g = i >> 6, j = i & 63;
  float v = pool[i] / fmaxf(cnt[g], 1.0f);
  xg[i] = v;                                 // d_out xg block  [G,64]
  comb[(size_t)g * 128 + j] = v;             // combined cols 0..63
}

// out[g] = c1[g,:] . Wc2 + bc2
__global__ void k_final_dot(const float* __restrict__ c1, const float* __restrict__ Wc2,
                            const float* __restrict__ bc2, float* __restrict__ outp, int G) {
  int g = blockIdx.x * blockDim.x + threadIdx.x;
  if (g >= G) return;
  float s = bc2[0];
  for (int k = 0; k < 128; ++k) s += c1[(size_t)g * 128 + k] * Wc2[k];
  outp[g] = s;
}

// ---------------------------------------------------------------------------
// Host-side orchestration
// ---------------------------------------------------------------------------
static void launch_gemm(const float* A, const float* B, const float* bias, float* C,
                        int M, int N, int K, int lda, int ldb, int ldc, int relu,
                        hipStream_t s) {
  int waves = (M >> 4) * (N >> 6);      // one wave per 16x64 strip
  gemm_f32_wmma<<<DIVUP(waves, 8), 256, 0, s>>>(A, B, bias, C, M, N, K, lda, ldb, ldc, relu);
}

static void run_gat(const float* X, int IN, int H, int C,
                    const float* W, const float* We, const float* a_s, const float* a_d,
                    const float* a_e, const float* bias,
                    const int* src, const int* dst, const float* ea,
                    float* bufH, float* outp,
                    float* aS, float* aD, float* alE, float* amax, float* denom, float* Mp,
                    hipStream_t s) {
  int OUT = H * C;
  launch_gemm(X, W, nullptr, bufH, Nn, OUT, IN, IN, OUT, OUT, 0, s);       // h = X@W
  k_alpha_nodes<<<DIVUP(Nn * H, 256), 256, 0, s>>>(bufH, a_s, a_d, aS, aD, Nn, H, C);
  k_mproj<<<1, 32, 0, s>>>(We, a_e, Mp, H, C);
  k_alpha_edge<<<DIVUP(Ee * H, 256), 256, 0, s>>>(ea, Mp, alE, Ee, H);
  k_init_out<<<DIVUP(Nn * OUT, 256), 256, 0, s>>>(outp, bias, Nn * OUT, OUT);
  k_init_nh<<<DIVUP(Nn * H, 256), 256, 0, s>>>(amax, denom, Nn * H);
  k_edge_lrelu_max<<<DIVUP(Ee * H, 256), 256, 0, s>>>(src, dst, aS, aD, alE, amax, Ee, H);
  k_edge_exp<<<DIVUP(Ee * H, 256), 256, 0, s>>>(dst, alE, amax, denom, Ee, H);
  k_edge_msg<<<Ee, OUT, 0, s>>>(src, dst, bufH, alE, denom, outp, H, C);
}

static void run_bn_relu(float* X, const float* g, const float* b, int OUT,
                        float* bnsum, float* bnsq, hipStream_t s) {
  k_zero<<<1, 256, 0, s>>>(bnsum, OUT);
  k_zero<<<1, 256, 0, s>>>(bnsq, OUT);
  k_bn_stats<<<DIVUP(Nn, 64), OUT, 0, s>>>(X, bnsum, bnsq, Nn, OUT, 64);
  k_bn_apply<<<DIVUP(Nn * OUT, 256), 256, 0, s>>>(X, bnsum, bnsq, g, b, Nn, OUT);
}

extern "C" void kernel_launch(void* const* d_in, const int* in_sizes, int n_in,
                              void* d_out, int out_size, void* d_ws, size_t ws_size,
                              hipStream_t stream) {
  const float* x    = (const float*)d_in[0];
  const int*  eidx  = (const int*)d_in[1];
  const float* ea   = (const float*)d_in[2];
  const int*  batch = (const int*)d_in[3];
  const float* exf  = (const float*)d_in[4];
  const float *W1 = (const float*)d_in[5],  *We1 = (const float*)d_in[6];
  const float *as1 = (const float*)d_in[7], *ad1 = (const float*)d_in[8];
  const float *ae1 = (const float*)d_in[9], *b1 = (const float*)d_in[10];
  const float *g1 = (const float*)d_in[11], *be1 = (const float*)d_in[12];
  const float *W2 = (const float*)d_in[13], *We2 = (const float*)d_in[14];
  const float *as2 = (const float*)d_in[15], *ad2 = (const float*)d_in[16];
  const float *ae2 = (const float*)d_in[17], *b2 = (const float*)d_in[18];
  const float *g2 = (const float*)d_in[19], *be2 = (const float*)d_in[20];
  const float *W3 = (const float*)d_in[21], *We3 = (const float*)d_in[22];
  const float *as3 = (const float*)d_in[23], *ad3 = (const float*)d_in[24];
  const float *ae3 = (const float*)d_in[25], *b3 = (const float*)d_in[26];
  const float *g3 = (const float*)d_in[27], *be3 = (const float*)d_in[28];
  const float *Wf1 = (const float*)d_in[29], *bf1 = (const float*)d_in[30];
  const float *Wf2 = (const float*)d_in[31], *bf2 = (const float*)d_in[32];
  const float *Wc1 = (const float*)d_in[33], *bc1 = (const float*)d_in[34];
  const float *Wc2 = (const float*)d_in[35], *bc2 = (const float*)d_in[36];

  const int* src = eidx;           // edge_index[0,:]
  const int* dst = eidx + Ee;      // edge_index[1,:]

  // workspace layout (offset allocator, 64-float alignment)
  float* ws = (float*)d_ws;
  size_t off = 0;
  auto alloc = [&](size_t n) -> float* {
    float* p = ws + off; off += (n + 63) & ~(size_t)63; return p;
  };
  float* bufH  = alloc((size_t)Nn * 256);   // linear output h
  float* bufP  = alloc((size_t)Nn * 256);   // gat1 out / gat3 out
  float* bufQ  = alloc((size_t)Nn * 256);   // gat2 out
  float* aS    = alloc((size_t)Nn * 4);
  float* aD    = alloc((size_t)Nn * 4);
  float* alE   = alloc((size_t)Ee * 4);     // alpha_e -> alpha -> ex (in place)
  float* amax  = alloc((size_t)Nn * 4);
  float* denom = alloc((size_t)Nn * 4);
  float* Mp    = alloc(64);
  float* bnsum = alloc(256);
  float* bnsq  = alloc(256);
  float* pool  = alloc((size_t)Gg * 64);
  float* cnt   = alloc(Gg);
  float* e1b   = alloc((size_t)Gg * 64);
  float* c1b   = alloc((size_t)Gg * 128);

  // d_out layout: (out[G,1] | xg[G,64] | combined[G,128]) flattened
  float* out_final = (float*)d_out;
  float* out_xg    = out_final + Gg;
  float* out_comb  = out_xg + (size_t)Gg * 64;

  // ---- GAT layer 1: 64 -> 4 heads x 64, concat; BN+ReLU ----
  run_gat(x, 64, 4, 64, W1, We1, as1, ad1, ae1, b1, src, dst, ea,
          bufH, bufP, aS, aD, alE, amax, denom, Mp, stream);
  run_bn_relu(bufP, g1, be1, 256, bnsum, bnsq, stream);

  // ---- GAT layer 2: 256 -> 4 heads x 64, concat; BN+ReLU ----
  run_gat(bufP, 256, 4, 64, W2, We2, as2, ad2, ae2, b2, src, dst, ea,
          bufH, bufQ, aS, aD, alE, amax, denom, Mp, stream);
  run_bn_relu(bufQ, g2, be2, 256, bnsum, bnsq, stream);

  // ---- GAT layer 3: 256 -> 1 head x 64 (mean over 1 head == identity); BN+ReLU ----
  run_gat(bufQ, 256, 1, 64, W3, We3, as3, ad3, ae3, b3, src, dst, ea,
          bufH, bufP, aS, aD, alE, amax, denom, Mp, stream);
  run_bn_relu(bufP, g3, be3, 64, bnsum, bnsq, stream);

  // ---- global mean pool over graphs ----
  k_zero<<<DIVUP(Gg * 64, 256), 256, 0, stream>>>(pool, Gg * 64);
  k_zero<<<DIVUP(Gg, 256), 256, 0, stream>>>(cnt, Gg);
  k_cnt<<<DIVUP(Nn, 256), 256, 0, stream>>>(batch, cnt, Nn);
  k_pool_scatter<<<DIVUP(Nn * 64, 256), 256, 0, stream>>>(bufP, batch, pool, Nn);
  k_pool_final<<<DIVUP(Gg * 64, 256), 256, 0, stream>>>(pool, cnt, out_xg, out_comb, Gg);

  // ---- experimental MLP branch (WMMA) ----
  launch_gemm(exf, Wf1, bf1, e1b, Gg, 64, 32, 32, 64, 64, 1, stream);            // relu
  launch_gemm(e1b, Wf2, bf2, out_comb + 64, Gg, 64, 64, 64, 64, 128, 1, stream); // relu -> combined[:,64:]

  // ---- fusion head ----
  launch_gemm(out_comb, Wc1, bc1, c1b, Gg, 128, 128, 128, 128, 128, 1, stream);  // relu
  k_final_dot<<<DIVUP(Gg, 128), 128, 0, stream>>>(c1b, Wc2, bc2, out_final, Gg);
}